// GLIF_84808424227210
// MI455X (gfx1250) — compile-verified
//
#include <hip/hip_runtime.h>
#include <math.h>

// GLIF spiking-neuron forward, MI455X (gfx1250).
// T=16, B=4096, C=1024 fp32. Pure streaming problem: 512MB traffic -> ~22us
// floor at 23.3 TB/s. No matmul content (spike threshold is nonlinear), so
// WMMA does not apply; we target the bandwidth roofline with:
//  - per-thread 16-step recurrence held in registers (float4 over channels)
//  - non-temporal B128 loads/stores for the 512MB stream (protect 192MB L2)
//  - CDNA5 async global->LDS DMA (ASYNCcnt path), 16 fully-unrolled
//    back-to-back issues staging the 64KB conduct-factor table per block
//  - all sigmoids precomputed once into workspace (no transcendentals in loop)

typedef __attribute__((ext_vector_type(4))) float v4f;

__device__ __forceinline__ float sigmoidf_(float v) {
    return 1.0f / (1.0f + __expf(-v));
}

// ---------------------------------------------------------------------------
// Prep kernel: fold all per-channel / per-(t,c) constants.
//   condf[t,c]   = 1 - be*(1 - sigmoid(conduct[t,c]))
//   decay[c]     = 1 - al*(1 - sigmoid(tau))
//   leakt[c]     = (1-al)*sigmoid(leak)
//   rev[c]       = (1-ga)*sigmoid(reVth)
//   vth[c]       = sigmoid(Vth)
//   gav[c]       = ga
// ---------------------------------------------------------------------------
__global__ void glif_prep(const float* __restrict__ tau,
                          const float* __restrict__ Vth,
                          const float* __restrict__ leak,
                          const float* __restrict__ reVth,
                          const float* __restrict__ conduct,
                          const float* __restrict__ alpha,
                          const float* __restrict__ beta,
                          const float* __restrict__ gamma,
                          float* __restrict__ condf,
                          float* __restrict__ decay,
                          float* __restrict__ leakt,
                          float* __restrict__ rev,
                          float* __restrict__ vth,
                          float* __restrict__ gav,
                          int T, int C)
{
    int idx = blockIdx.x * blockDim.x + threadIdx.x;
    int TC = T * C;
    if (idx < TC) {
        int c = idx % C;
        float be = (beta[c] > 0.0f) ? 1.0f : 0.0f;
        float cs = sigmoidf_(conduct[idx]);
        condf[idx] = 1.0f - be * (1.0f - cs);
    }
    if (idx < C) {
        float al = (alpha[idx] > 0.0f) ? 1.0f : 0.0f;
        float ga = (gamma[idx] > 0.0f) ? 1.0f : 0.0f;
        decay[idx] = 1.0f - al * (1.0f - sigmoidf_(tau[idx]));
        leakt[idx] = (1.0f - al) * sigmoidf_(leak[idx]);
        rev[idx]   = (1.0f - ga) * sigmoidf_(reVth[idx]);
        vth[idx]   = sigmoidf_(Vth[idx]);
        gav[idx]   = ga;
    }
}

// ---------------------------------------------------------------------------
// Main kernel. One block per batch row b (C channels, BLOCK threads x float4,
// BLOCK == C/4). condf (T*C floats = 64KB) is staged into LDS with 16
// fully-unrolled CDNA5 async-DMA issues + one ASYNCcnt wait.
// ---------------------------------------------------------------------------
template <int T, int C, int BLOCK>
__global__ __launch_bounds__(BLOCK)
void glif_main(const float* __restrict__ x,
               const float* __restrict__ condf,
               const float* __restrict__ decay,
               const float* __restrict__ leakt,
               const float* __restrict__ rev,
               const float* __restrict__ vth,
               const float* __restrict__ gav,
               float* __restrict__ out,
               int B)
{
    static_assert(BLOCK * 4 == C, "one block covers one full channel row");
    __shared__ float s_condf[T * C];      // 64KB for T=16, C=1024

    const int tid = threadIdx.x;
    const int b   = blockIdx.x;
    const int c4  = tid * 4;              // first of 4 contiguous channels

    // ---- async-stage condf[T*C] into LDS: 16 back-to-back B128 DMA issues ---
    {
        // low 32 bits of a generic LDS pointer are the LDS byte address
        unsigned lds_base = (unsigned)(size_t)(&s_condf[0]) + (unsigned)(tid * 16);
        const char* gbase = (const char*)condf + tid * 16;
        constexpr int ITERS = (T * C * 4) / (BLOCK * 16);   // = 16
#pragma unroll
        for (int i = 0; i < ITERS; ++i) {
            unsigned laddr = lds_base + (unsigned)(i * BLOCK * 16);
            const char* gaddr = gbase + i * BLOCK * 16;
            asm volatile("global_load_async_to_lds_b128 %0, %1, off"
                         :: "v"(laddr), "v"(gaddr)
                         : "memory");
        }
        asm volatile("s_wait_asynccnt 0" ::: "memory");
    }
    __syncthreads();

    // ---- per-channel constants (tiny tables, L2-resident, regular loads) ----
    const v4f dec = *(const v4f*)(decay + c4);
    const v4f lkt = *(const v4f*)(leakt + c4);
    const v4f rv  = *(const v4f*)(rev   + c4);
    const v4f vt  = *(const v4f*)(vth   + c4);
    const v4f ga  = *(const v4f*)(gav   + c4);
    const v4f one = {1.0f, 1.0f, 1.0f, 1.0f};

    // ---- issue all 16 streaming loads up front (independent of recurrence) --
    const size_t plane = (size_t)B * (size_t)C;       // elements per time step
    const size_t basex = (size_t)b * (size_t)C + (size_t)c4;

    v4f xv[T];
#pragma unroll
    for (int t = 0; t < T; ++t) {
        xv[t] = __builtin_nontemporal_load(
                    (const v4f*)(x + (size_t)t * plane + basex));
    }

    // ---- 16-step recurrence entirely in registers ----
    v4f u = {0.0f, 0.0f, 0.0f, 0.0f};
    v4f o = {0.0f, 0.0f, 0.0f, 0.0f};
#pragma unroll
    for (int t = 0; t < T; ++t) {
        v4f cf = *(const v4f*)(&s_condf[t * C + c4]);
        v4f I  = xv[t] * cf;
        u = dec * u * (one - ga * o) - lkt + I - rv * o;
        v4f d = u - vt;
        o.x = (d.x >= 0.0f) ? 1.0f : 0.0f;
        o.y = (d.y >= 0.0f) ? 1.0f : 0.0f;
        o.z = (d.z >= 0.0f) ? 1.0f : 0.0f;
        o.w = (d.w >= 0.0f) ? 1.0f : 0.0f;
        __builtin_nontemporal_store(o, (v4f*)(out + (size_t)t * plane + basex));
    }
}

// ---------------------------------------------------------------------------
extern "C" void kernel_launch(void* const* d_in, const int* in_sizes, int n_in,
                              void* d_out, int out_size, void* d_ws, size_t ws_size,
                              hipStream_t stream)
{
    const float* x       = (const float*)d_in[0];
    const float* tau     = (const float*)d_in[1];
    const float* Vth     = (const float*)d_in[2];
    const float* leak    = (const float*)d_in[3];
    const float* reVth   = (const float*)d_in[4];
    const float* conduct = (const float*)d_in[5];
    const float* alpha   = (const float*)d_in[6];
    const float* beta    = (const float*)d_in[7];
    const float* gamma   = (const float*)d_in[8];

    const int C = in_sizes[1];                  // 1024
    const int T = in_sizes[5] / C;              // 16
    const int B = (int)((long long)in_sizes[0] / ((long long)T * C)); // 4096

    // workspace layout: condf[T*C] | decay[C] | leakt[C] | rev[C] | vth[C] | gav[C]
    float* ws    = (float*)d_ws;
    float* condf = ws;
    float* decay = condf + (size_t)T * C;
    float* leakt = decay + C;
    float* rev   = leakt + C;
    float* vth   = rev + C;
    float* gav   = vth + C;

    int prep_n = T * C;
    glif_prep<<<(prep_n + 255) / 256, 256, 0, stream>>>(
        tau, Vth, leak, reVth, conduct, alpha, beta, gamma,
        condf, decay, leakt, rev, vth, gav, T, C);

    // one block per batch row; C/4 threads (=256 for C=1024, 8 waves of 32)
    glif_main<16, 1024, 256><<<dim3(B), dim3(256), 0, stream>>>(
        x, condf, decay, leakt, rev, vth, gav, (float*)d_out, B);
}